// PointNetFeaturePropogation_43739946942748
// MI455X (gfx1250) — compile-verified
//
#include <hip/hip_runtime.h>

typedef __attribute__((ext_vector_type(2))) float v2f;
typedef __attribute__((ext_vector_type(8))) float v8f;

#define B_   4
#define N_   8192
#define S_   2048
#define D1_  128
#define D2_  256
#define C1_  384      // D1_+D2_
#define H_   256
#define LDSTR 40      // padded LDS row stride (floats): half-wave K rows hit disjoint banks

// ---------------- three_nn + inverse-distance weights ----------------
__global__ void k_three_nn(const float* __restrict__ pos1,
                           const float* __restrict__ pos2,
                           int* __restrict__ idx, float* __restrict__ wgt) {
  __shared__ float sx[S_], sy[S_], sz[S_];
  const int b = blockIdx.y;
  const int n = blockIdx.x * 256 + threadIdx.x;
  const float* p2 = pos2 + (size_t)b * 3 * S_;
  for (int s = threadIdx.x; s < S_; s += 256) {
    sx[s] = p2[s];
    sy[s] = p2[S_ + s];
    sz[s] = p2[2 * S_ + s];
  }
  __syncthreads();
  const float px = pos1[(size_t)b * 3 * N_ + n];
  const float py = pos1[(size_t)b * 3 * N_ + N_ + n];
  const float pz = pos1[(size_t)b * 3 * N_ + 2 * N_ + n];
  float d0 = 3.4e38f, d1 = 3.4e38f, d2 = 3.4e38f;
  int   i0 = 0, i1 = 0, i2 = 0;
  for (int s = 0; s < S_; ++s) {
    float dx = px - sx[s], dy = py - sy[s], dz = pz - sz[s];
    float d = dx * dx + dy * dy + dz * dz;
    if (d < d0)      { d2 = d1; i2 = i1; d1 = d0; i1 = i0; d0 = d; i0 = s; }
    else if (d < d1) { d2 = d1; i2 = i1; d1 = d;  i1 = s; }
    else if (d < d2) { d2 = d;  i2 = s; }
  }
  d0 = fmaxf(d0, 1e-10f); d1 = fmaxf(d1, 1e-10f); d2 = fmaxf(d2, 1e-10f);
  float w0 = 1.f / d0, w1 = 1.f / d1, w2 = 1.f / d2;
  float inv = 1.f / (w0 + w1 + w2);
  size_t o = ((size_t)b * N_ + n) * 3;
  idx[o] = i0; idx[o + 1] = i1; idx[o + 2] = i2;
  wgt[o] = w0 * inv; wgt[o + 1] = w1 * inv; wgt[o + 2] = w2 * inv;
}

// ---------------- GEMM1: fused interpolation+concat loader, f32 WMMA ----------------
// block: 256 threads = 8 waves; tile: 64 out-ch x 32 points; wave -> one 16x16 tile
__global__ void k_gemm1(const float* __restrict__ W, const float* __restrict__ bias,
                        const float* __restrict__ feature1, const float* __restrict__ feature2,
                        const int* __restrict__ idx, const float* __restrict__ wgt,
                        float* __restrict__ Y) {
  extern __shared__ float lds[];        // C1_ x LDSTR floats
  __shared__ int   sIdx[32 * 3];
  __shared__ float sW[32 * 3];
  const int tid = threadIdx.x;
  const int b  = blockIdx.z;
  const int n0 = blockIdx.x * 32;
  const int mB = blockIdx.y * 64;
  if (tid < 96) {
    size_t o = ((size_t)b * N_ + n0) * 3 + tid;
    sIdx[tid] = idx[o];
    sW[tid]   = wgt[o];
  }
  __syncthreads();
  // build the 384x32 input slab: rows 0..255 interpolated feature2, rows 256..383 feature1
  for (int e = tid; e < C1_ * 32; e += 256) {
    int c = e >> 5, j = e & 31;
    float v;
    if (c < D2_) {
      const float* f2 = feature2 + ((size_t)b * D2_ + c) * S_;
      v = sW[j * 3 + 0] * f2[sIdx[j * 3 + 0]]
        + sW[j * 3 + 1] * f2[sIdx[j * 3 + 1]]
        + sW[j * 3 + 2] * f2[sIdx[j * 3 + 2]];
    } else {
      v = feature1[((size_t)b * D1_ + (c - D2_)) * N_ + n0 + j];
    }
    lds[c * LDSTR + j] = v;
  }
  __syncthreads();
  const int lane  = tid & 31, wave = tid >> 5;
  const int mtile = wave >> 1, ntile = wave & 1;
  const int khalf = (lane >> 4) << 1;                 // upper half-wave holds K+2
  const int m     = mB + mtile * 16 + (lane & 15);
  const int ncol  = ntile * 16 + (lane & 15);
  v8f acc = {};
#pragma unroll 4
  for (int kb = 0; kb < C1_; kb += 4) {
    v2f a = *(const v2f*)(W + (size_t)m * C1_ + kb + khalf);
    v2f bb;
    bb.x = lds[(kb + khalf) * LDSTR + ncol];
    bb.y = lds[(kb + khalf + 1) * LDSTR + ncol];
    acc = __builtin_amdgcn_wmma_f32_16x16x4_f32(false, a, false, bb,
                                                (short)0, acc, false, false);
  }
  const int mlo  = mB + mtile * 16 + ((lane >> 4) << 3);
  const int nout = n0 + ntile * 16 + (lane & 15);
#pragma unroll
  for (int r = 0; r < 8; ++r) {
    int mo = mlo + r;
    Y[((size_t)b * H_ + mo) * N_ + nout] = acc[r] + bias[mo];
  }
}

// ---------------- GEMM2: loader applies BN1+ReLU on the fly ----------------
__global__ void k_gemm2(const float* __restrict__ W, const float* __restrict__ bias,
                        const float* __restrict__ Yin, const float* __restrict__ scale,
                        const float* __restrict__ shift, float* __restrict__ Y) {
  extern __shared__ float lds[];        // H_ x LDSTR floats
  const int tid = threadIdx.x;
  const int b  = blockIdx.z;
  const int n0 = blockIdx.x * 32;
  const int mB = blockIdx.y * 64;
  for (int e = tid; e < H_ * 32; e += 256) {
    int c = e >> 5, j = e & 31;
    float v = Yin[((size_t)b * H_ + c) * N_ + n0 + j];
    lds[c * LDSTR + j] = fmaxf(v * scale[c] + shift[c], 0.f);
  }
  __syncthreads();
  const int lane  = tid & 31, wave = tid >> 5;
  const int mtile = wave >> 1, ntile = wave & 1;
  const int khalf = (lane >> 4) << 1;
  const int m     = mB + mtile * 16 + (lane & 15);
  const int ncol  = ntile * 16 + (lane & 15);
  v8f acc = {};
#pragma unroll 4
  for (int kb = 0; kb < H_; kb += 4) {
    v2f a = *(const v2f*)(W + (size_t)m * H_ + kb + khalf);
    v2f bb;
    bb.x = lds[(kb + khalf) * LDSTR + ncol];
    bb.y = lds[(kb + khalf + 1) * LDSTR + ncol];
    acc = __builtin_amdgcn_wmma_f32_16x16x4_f32(false, a, false, bb,
                                                (short)0, acc, false, false);
  }
  const int mlo  = mB + mtile * 16 + ((lane >> 4) << 3);
  const int nout = n0 + ntile * 16 + (lane & 15);
#pragma unroll
  for (int r = 0; r < 8; ++r) {
    int mo = mlo + r;
    Y[((size_t)b * H_ + mo) * N_ + nout] = acc[r] + bias[mo];
  }
}

// ---------------- per-channel batch stats -> (scale, shift) ----------------
__global__ void k_bn_stats(const float* __restrict__ Y, const float* __restrict__ gamma,
                           const float* __restrict__ beta, float* __restrict__ scale,
                           float* __restrict__ shift) {
  __shared__ float rs[256], rq[256];
  const int c = blockIdx.x, tid = threadIdx.x;
  float s = 0.f, q = 0.f;
  for (int b = 0; b < B_; ++b) {
    const float* p = Y + ((size_t)b * H_ + c) * N_;
    for (int n = tid; n < N_; n += 256) { float v = p[n]; s += v; q += v * v; }
  }
  rs[tid] = s; rq[tid] = q;
  __syncthreads();
  for (int st = 128; st > 0; st >>= 1) {
    if (tid < st) { rs[tid] += rs[tid + st]; rq[tid] += rq[tid + st]; }
    __syncthreads();
  }
  if (tid == 0) {
    const float invM = 1.f / (float)(B_ * N_);
    float mean = rs[0] * invM;
    float var  = rq[0] * invM - mean * mean;
    float sc   = gamma[c] * rsqrtf(var + 1e-5f);
    scale[c] = sc;
    shift[c] = beta[c] - mean * sc;
  }
}

// ---------------- final BN2 + ReLU, in place on d_out ----------------
__global__ void k_bn_relu(float* __restrict__ Y, const float* __restrict__ scale,
                          const float* __restrict__ shift) {
  size_t i = (size_t)blockIdx.x * 256 + threadIdx.x;
  int c = (int)((i >> 13) & (H_ - 1));     // N_ = 8192 = 2^13
  float v = Y[i];
  Y[i] = fmaxf(v * scale[c] + shift[c], 0.f);
}

extern "C" void kernel_launch(void* const* d_in, const int* in_sizes, int n_in,
                              void* d_out, int out_size, void* d_ws, size_t ws_size,
                              hipStream_t stream) {
  const float* pos1     = (const float*)d_in[0];
  const float* pos2     = (const float*)d_in[1];
  const float* feature1 = (const float*)d_in[2];
  const float* feature2 = (const float*)d_in[3];
  const float* W1  = (const float*)d_in[4];
  const float* b1  = (const float*)d_in[5];
  const float* g1  = (const float*)d_in[6];
  const float* be1 = (const float*)d_in[7];
  const float* W2  = (const float*)d_in[8];
  const float* b2  = (const float*)d_in[9];
  const float* g2  = (const float*)d_in[10];
  const float* be2 = (const float*)d_in[11];

  char* ws = (char*)d_ws;
  int*   idx = (int*)ws;        ws += (size_t)B_ * N_ * 3 * sizeof(int);
  float* wgt = (float*)ws;      ws += (size_t)B_ * N_ * 3 * sizeof(float);
  float* Y1  = (float*)ws;      ws += (size_t)B_ * H_ * N_ * sizeof(float);
  float* scale1 = (float*)ws;   ws += H_ * sizeof(float);
  float* shift1 = (float*)ws;   ws += H_ * sizeof(float);
  float* scale2 = (float*)ws;   ws += H_ * sizeof(float);
  float* shift2 = (float*)ws;   ws += H_ * sizeof(float);
  float* out = (float*)d_out;

  k_three_nn<<<dim3(N_ / 256, B_), 256, 0, stream>>>(pos1, pos2, idx, wgt);

  k_gemm1<<<dim3(N_ / 32, H_ / 64, B_), 256, C1_ * LDSTR * sizeof(float), stream>>>(
      W1, b1, feature1, feature2, idx, wgt, Y1);

  k_bn_stats<<<H_, 256, 0, stream>>>(Y1, g1, be1, scale1, shift1);

  k_gemm2<<<dim3(N_ / 32, H_ / 64, B_), 256, H_ * LDSTR * sizeof(float), stream>>>(
      W2, b2, Y1, scale1, shift1, out);

  k_bn_stats<<<H_, 256, 0, stream>>>(out, g2, be2, scale2, shift2);

  k_bn_relu<<<(B_ * H_ * N_) / 256, 256, 0, stream>>>(out, scale2, shift2);
}